// GAT_54932631715853
// MI455X (gfx1250) — compile-verified
//
#include <hip/hip_runtime.h>
#include <cstdint>

#define N_NODES   4096
#define NODE_DIMC 256
#define HIDDENC   64
#define NHEADS    4
#define NCLASSES  16
#define NEG_SLOPE 0.01f

typedef __attribute__((ext_vector_type(2))) float v2f;
typedef __attribute__((ext_vector_type(8))) float v8f;

// ---- persistent scratch (zero-init .bss; every element rewritten each call) ----
__device__ __align__(16) float    g_H1[NHEADS * N_NODES * HIDDENC];     // 4 MB
__device__ __align__(16) float    g_F1[NHEADS * N_NODES];
__device__ __align__(16) float    g_F2[NHEADS * N_NODES];
__device__ __align__(16) uint32_t g_adj[N_NODES * (N_NODES / 32)];      // 2 MB bitmask
__device__ __align__(16) float    g_Hcat[N_NODES * (NHEADS * HIDDENC)]; // 4 MB
__device__ __align__(16) float    g_H2[N_NODES * NCLASSES];
__device__ __align__(16) float    g_F1b[N_NODES];
__device__ __align__(16) float    g_F2b[N_NODES];

// ---- pack adjacency into bitmask: one wave produces one u32 word ----
__global__ void pack_adj_kernel(const int* __restrict__ adj) {
  unsigned tid = blockIdx.x * blockDim.x + threadIdx.x;   // i*4096 + j
  int v = adj[tid] != 0;
#if __has_builtin(__builtin_amdgcn_ballot_w32)
  uint32_t m = __builtin_amdgcn_ballot_w32(v);
  if ((tid & 31u) == 0u) g_adj[tid >> 5] = m;
#else
  if (v) atomicOr(&g_adj[tid >> 5], 1u << (tid & 31u));   // idempotent fallback
#endif
}

// ---- layer-1 features H1[h] = x @ W1[h]^T via V_WMMA_F32_16X16X4_F32 ----
__global__ __launch_bounds__(128) void feat1_kernel(const float* __restrict__ x,
                                                    const float* __restrict__ W1) {
  const int head  = blockIdx.x >> 8;
  const int itile = blockIdx.x & 255;
  const int wave  = threadIdx.x >> 5;      // o-tile 0..3
  const int lane  = threadIdx.x & 31;
  const int r16   = lane & 15;
  const int khalf = (lane < 16) ? 0 : 2;   // ISA A/B lane-half K split
  const float* xr = x  + (size_t)(itile * 16 + r16) * NODE_DIMC;
  const float* wr = W1 + (size_t)(head * HIDDENC + wave * 16 + r16) * NODE_DIMC;
  v8f c = {};
  for (int k = 0; k < NODE_DIMC; k += 4) {
    v2f a = *(const v2f*)(xr + k + khalf);   // A: M=lane%16, K=khalf+{0,1}
    v2f b = *(const v2f*)(wr + k + khalf);   // B: N=lane%16, K=khalf+{0,1}
    c = __builtin_amdgcn_wmma_f32_16x16x4_f32(false, a, false, b, (short)0, c,
                                              false, false);
  }
  float* out = g_H1 + (size_t)head * N_NODES * HIDDENC;
  const int rbase = itile * 16 + ((lane < 16) ? 0 : 8);
  const int col   = wave * 16 + r16;
#pragma unroll
  for (int r = 0; r < 8; ++r)
    out[(size_t)(rbase + r) * HIDDENC + col] = c[r];
}

// ---- f1/f2 attention vectors, layer 1 ----
__global__ void fvec1_kernel(const float* __restrict__ a1, const float* __restrict__ a2) {
  int id = blockIdx.x * blockDim.x + threadIdx.x;   // head*4096 + i
  int head = id >> 12;
  const float* hr  = g_H1 + (size_t)id * HIDDENC;
  const float* a1h = a1 + head * HIDDENC;
  const float* a2h = a2 + head * HIDDENC;
  float s1 = 0.f, s2 = 0.f;
#pragma unroll 8
  for (int o = 0; o < HIDDENC; ++o) {
    float v = hr[o]; s1 = fmaf(v, a1h[o], s1); s2 = fmaf(v, a2h[o], s2);
  }
  g_F1[id] = s1; g_F2[id] = s2;
}

// ---- fused GAT attention layer: 16 rows per block, flash-style ----
template<int DIM, bool ELU_OUT>
__global__ __launch_bounds__(128) void attn_kernel(float* __restrict__ out2) {
  const float* F1g = (DIM == HIDDENC) ? (const float*)g_F1 : (const float*)g_F1b;
  const float* F2g = (DIM == HIDDENC) ? (const float*)g_F2 : (const float*)g_F2b;
  const float* Vg  = (DIM == HIDDENC) ? (const float*)g_H1 : (const float*)g_H2;
  float* outp      = (DIM == HIDDENC) ? (float*)g_Hcat : out2;
  const int ostride = (DIM == HIDDENC) ? (NHEADS * HIDDENC) : NCLASSES;

  constexpr int NT  = DIM / 16;   // N-tiles (4 for layer1, 1 for layer2)
  constexpr int KSL = 4 / NT;     // K-slices per N-tile (1 or 4)

  __shared__ __align__(16) float    sm_f2[N_NODES];     // 16 KB
  __shared__ __align__(16) uint32_t sm_adj[16 * 128];   // 8 KB
  __shared__ __align__(16) float    sA[16 * 128];       // 8 KB staged exp weights
  __shared__ float sm_f1[16], sm_m[16], sm_d[16];
  __shared__ float sm_red[128];
  __shared__ float sm_acc[16 * 16];

  const int head  = blockIdx.x >> 8;                    // 0 for layer 2
  const int itile = blockIdx.x & 255;
  const int i0    = itile * 16;
  const int t     = threadIdx.x;
  const float* f2 = F2g + (size_t)head * N_NODES;
  const float* V  = Vg  + (size_t)head * N_NODES * DIM;

  for (int idx = t; idx < N_NODES / 4; idx += 128)
    ((float4*)sm_f2)[idx] = ((const float4*)f2)[idx];
  for (int idx = t; idx < (16 * 128) / 4; idx += 128)
    ((uint4*)sm_adj)[idx] = ((const uint4*)(g_adj + (size_t)i0 * 128))[idx];
  if (t < 16) { sm_f1[t] = F1g[(size_t)head * N_NODES + i0 + t]; sm_d[t] = 0.f; }
  if (NT == 1) for (int idx = t; idx < 256; idx += 128) sm_acc[idx] = 0.f;
  __syncthreads();

  const int row = t & 15;   // staging row assignment (8 threads per row)
  const int grp = t >> 4;

  // ---- pass 1: masked row maxima (leaky-relu is monotone -> max of raw) ----
  {
    const float f1r = sm_f1[row];
    const uint32_t* ar = sm_adj + row * 128;
    float rmax = -__builtin_inff();
    for (int j = grp; j < N_NODES; j += 8) {
      float raw = f1r + sm_f2[j];
      bool vld = ((ar[j >> 5] >> (j & 31)) & 1u) && (raw != 0.f);
      rmax = vld ? fmaxf(rmax, raw) : rmax;
    }
    sm_red[t] = rmax;
  }
  __syncthreads();
  if (t < 16) {
    float m = -__builtin_inff();
#pragma unroll
    for (int g = 0; g < 8; ++g) m = fmaxf(m, sm_red[t + 16 * g]);
    sm_m[t] = (m > 0.f) ? m : NEG_SLOPE * m;   // lrelu(max raw)
  }
  __syncthreads();

  // ---- pass 2: stage exp weights in LDS, accumulate with f32 WMMA ----
  const int wave   = t >> 5;
  const int lane   = t & 31;
  const int ntile  = wave % NT;
  const int kslice = wave / NT;
  const int khalf  = (lane < 16) ? 0 : 2;
  const float* Vc  = V + ntile * 16 + (lane & 15);
  v8f acc = {};
  const float f1r = sm_f1[row];
  const float mr  = sm_m[row];
  const uint32_t* ar = sm_adj + row * 128;

  for (int j0 = 0; j0 < N_NODES; j0 += 128) {
    float dsum = 0.f;
    const int jb = j0 + grp * 16;
#pragma unroll
    for (int u = 0; u < 16; ++u) {
      const int j = jb + u;
      float raw = f1r + sm_f2[j];
      float lr  = (raw > 0.f) ? raw : NEG_SLOPE * raw;
      bool vld  = ((ar[j >> 5] >> (j & 31)) & 1u) && (raw != 0.f);
      float s   = vld ? __expf(lr - mr) : 0.f;
      sA[row * 128 + grp * 16 + u] = s;
      dsum += s;
    }
    atomicAdd(&sm_d[row], dsum);          // ds_add_f32: builds denominator online
    __syncthreads();

    const int s_begin = kslice * (32 / KSL);
#pragma unroll 8
    for (int st = s_begin; st < s_begin + (32 / KSL); ++st) {
      const int jj = st * 4 + khalf;
      v2f a, b;
      a.x = sA[(lane & 15) * 128 + jj];
      a.y = sA[(lane & 15) * 128 + jj + 1];
      b.x = Vc[(size_t)(j0 + jj) * DIM];
      b.y = Vc[(size_t)(j0 + jj + 1) * DIM];
      acc = __builtin_amdgcn_wmma_f32_16x16x4_f32(false, a, false, b, (short)0,
                                                  acc, false, false);
    }
    __syncthreads();
  }

  // ---- normalize + write ----
  const int ocol  = head * DIM;
  const int rbase = (lane < 16) ? 0 : 8;
  if (NT == 4) {                           // layer 1: each wave owns 16 cols
#pragma unroll
    for (int r = 0; r < 8; ++r) {
      const int rr = rbase + r;
      float o = acc[r] / sm_d[rr];
      if (ELU_OUT) o = (o > 0.f) ? o : expm1f(o);
      outp[(size_t)(i0 + rr) * ostride + ocol + (lane & 15)] = o;
    }
  } else {                                 // layer 2: reduce 4 K-slice waves
#pragma unroll
    for (int r = 0; r < 8; ++r)
      atomicAdd(&sm_acc[(rbase + r) * 16 + (lane & 15)], acc[r]);
    __syncthreads();
    for (int idx = t; idx < 256; idx += 128) {
      float o = sm_acc[idx] / sm_d[idx >> 4];
      if (ELU_OUT) o = (o > 0.f) ? o : expm1f(o);
      outp[(size_t)(i0 + (idx >> 4)) * ostride + ocol + (idx & 15)] = o;
    }
  }
}

// ---- layer-2 features H2 = elu(Hcat) @ W2^T (small: 16.7M MACs, VALU) ----
__global__ void feat2_kernel(const float* __restrict__ W2) {
  int id = blockIdx.x * blockDim.x + threadIdx.x;   // i*16 + c
  int i = id >> 4, c = id & 15;
  const float4* xr = (const float4*)(g_Hcat + (size_t)i * NODE_DIMC);
  const float4* wr = (const float4*)(W2 + (size_t)c * NODE_DIMC);
  float s = 0.f;
#pragma unroll 8
  for (int k = 0; k < NODE_DIMC / 4; ++k) {
    float4 xv = xr[k], wv = wr[k];
    s = fmaf(xv.x, wv.x, fmaf(xv.y, wv.y, fmaf(xv.z, wv.z, fmaf(xv.w, wv.w, s))));
  }
  g_H2[id] = s;
}

__global__ void fvec2_kernel(const float* __restrict__ a1, const float* __restrict__ a2) {
  int i = blockIdx.x * blockDim.x + threadIdx.x;
  const float* hr = g_H2 + (size_t)i * NCLASSES;
  float s1 = 0.f, s2 = 0.f;
#pragma unroll
  for (int c = 0; c < NCLASSES; ++c) {
    float v = hr[c]; s1 = fmaf(v, a1[c], s1); s2 = fmaf(v, a2[c], s2);
  }
  g_F1b[i] = s1; g_F2b[i] = s2;
}

extern "C" void kernel_launch(void* const* d_in, const int* in_sizes, int n_in,
                              void* d_out, int out_size, void* d_ws, size_t ws_size,
                              hipStream_t stream) {
  (void)in_sizes; (void)n_in; (void)out_size; (void)d_ws; (void)ws_size;
  const float* x    = (const float*)d_in[0];
  const int*   adj  = (const int*)d_in[1];
  const float* W1   = (const float*)d_in[2];
  const float* a1_1 = (const float*)d_in[3];
  const float* a2_1 = (const float*)d_in[4];
  const float* W2   = (const float*)d_in[5];
  const float* a1_2 = (const float*)d_in[6];
  const float* a2_2 = (const float*)d_in[7];
  float* out = (float*)d_out;

  pack_adj_kernel<<<(N_NODES * (size_t)N_NODES) / 256, 256, 0, stream>>>(adj);
  feat1_kernel<<<NHEADS * (N_NODES / 16), 128, 0, stream>>>(x, W1);
  fvec1_kernel<<<(NHEADS * N_NODES) / 256, 256, 0, stream>>>(a1_1, a2_1);
  attn_kernel<HIDDENC, true><<<NHEADS * (N_NODES / 16), 128, 0, stream>>>(nullptr);
  feat2_kernel<<<(N_NODES * NCLASSES) / 256, 256, 0, stream>>>(W2);
  fvec2_kernel<<<N_NODES / 256, 256, 0, stream>>>(a1_2, a2_2);
  attn_kernel<NCLASSES, false><<<N_NODES / 16, 128, 0, stream>>>(out);
}